// EfConv_15504831938608
// MI455X (gfx1250) — compile-verified
//
#include <hip/hip_runtime.h>

// EfConv forward, algebraically re-associated:
//   z   = node_feat @ W^T                  (WMMA f32 16x16x4, fp32-exact)
//   out[n, k*64+o] = b[o] + sum_{e:dst[e]=n} edge_feat[e,k] * z[src[e], o]
// Removes the 102MB agg intermediate plus the 8x-larger second GEMM.
//
// NOTE: efconv_scatter is placed FIRST in this file so the disasm snippet
// exposes the atomic lowering (want native global_atomic_add_f32 SCOPE_DEV,
// not a CAS loop).

typedef __attribute__((ext_vector_type(2))) float v2f;
typedef __attribute__((ext_vector_type(8))) float v8f;

#define IN_FEATS 64
#define OUT_FEATS 64
#define EDGE_DIM 8
#define OUT_STRIDE (EDGE_DIM * OUT_FEATS) // 512

// ---------------------------------------------------------------------------
// Edge scatter (the bottleneck phase). One wave32 per edge (grid-stride);
// each lane owns feature pair {2*lane, 2*lane+1}. out (102.4 MB) fits in the
// 192MB global L2, so the fp32 atomics resolve at L2 atomic throughput,
// never HBM. Atomics are non-returning (result discarded -> TH no-return),
// relaxed ordering, agent scope.
// ---------------------------------------------------------------------------
__global__ void __launch_bounds__(256)
efconv_scatter(const float* __restrict__ z, const float* __restrict__ ef,
               const int* __restrict__ src, const int* __restrict__ dst,
               float* __restrict__ out, int n_edges) {
  const int gwave  = (int)((blockIdx.x * blockDim.x + threadIdx.x) >> 5);
  const int nwaves = (int)((gridDim.x * blockDim.x) >> 5);
  const int lane   = (int)(threadIdx.x & 31);

  for (int e = gwave; e < n_edges; e += nwaves) {
    // Prefetch next iteration's edge-feature line (global_prefetch_b8).
    if (e + nwaves < n_edges) {
      __builtin_prefetch(ef + (size_t)(e + nwaves) * EDGE_DIM, 0, 0);
    }

    const int s = src[e];
    const int d = dst[e];

    // z row for this edge's source node: aligned 8B per lane, 256B/wave coalesced
    v2f h = *(const v2f*)(z + (size_t)s * OUT_FEATS + lane * 2);

    // 8 edge weights, broadcast (same address all lanes -> L0 broadcast)
    const float4* e4 = (const float4*)(ef + (size_t)e * EDGE_DIM);
    float4 ea = e4[0], eb = e4[1];
    float ew[EDGE_DIM] = {ea.x, ea.y, ea.z, ea.w, eb.x, eb.y, eb.z, eb.w};

    float* ob = out + (size_t)d * OUT_STRIDE + lane * 2;
#pragma unroll
    for (int k = 0; k < EDGE_DIM; ++k) {
      __hip_atomic_fetch_add(ob + k * OUT_FEATS + 0, h.x * ew[k],
                             __ATOMIC_RELAXED, __HIP_MEMORY_SCOPE_AGENT);
      __hip_atomic_fetch_add(ob + k * OUT_FEATS + 1, h.y * ew[k],
                             __ATOMIC_RELAXED, __HIP_MEMORY_SCOPE_AGENT);
    }
  }
}

// ---------------------------------------------------------------------------
// z[m, o] = sum_i node_feat[m, i] * W[o, i]
// One wave32 computes one 16x16 C tile via 16 x V_WMMA_F32_16X16X4_F32.
// A = node_feat tile (16xK, row-major), B[k][n] = W[n][k]  (both fp32).
// Requires n_nodes % 16 == 0 (50000 = 16 * 3125).
// ---------------------------------------------------------------------------
__global__ void __launch_bounds__(256)
efconv_gemm_zw(const float* __restrict__ nf, const float* __restrict__ W,
               float* __restrict__ z, int n_nodes) {
  const int wave  = (int)((blockIdx.x * blockDim.x + threadIdx.x) >> 5);
  const int lane  = (int)(threadIdx.x & 31);
  const int NT_N  = OUT_FEATS / 16;                 // 4 tiles along o
  const int total = (n_nodes / 16) * NT_N;
  if (wave >= total) return;                        // wave-uniform exit (EXEC all-1 inside)

  const int tm   = wave / NT_N;
  const int tn   = wave % NT_N;
  const int half = lane >> 4;                       // 0: lanes 0-15, 1: lanes 16-31
  const int l16  = lane & 15;

  // ISA 16x4 f32 A layout: lane L holds A[L%16][2*half + {0,1}] in v[0],v[1]
  const float* arow = nf + (size_t)(tm * 16 + l16) * IN_FEATS;      // M row
  // B (4x16) symmetric layout: lane L holds B[2*half+{0,1}][l16]; B[k][n] = W[n][k]
  const float* brow = W + (size_t)(tn * 16 + l16) * IN_FEATS;       // N row of W

  v8f c = {};
#pragma unroll
  for (int k0 = 0; k0 < IN_FEATS; k0 += 4) {
    v2f a = *(const v2f*)(arow + k0 + 2 * half);    // aligned global_load_b64
    v2f b = *(const v2f*)(brow + k0 + 2 * half);
    // 8 args: (neg_a, A, neg_b, B, c_mod, C, reuse_a, reuse_b)
    c = __builtin_amdgcn_wmma_f32_16x16x4_f32(false, a, false, b, (short)0, c,
                                              false, false);
  }

  // C/D layout: VGPR v -> row (v + 8*half), col l16
  float* zt = z + (size_t)(tm * 16) * OUT_FEATS + tn * 16 + l16;
#pragma unroll
  for (int v = 0; v < 8; ++v) {
    zt[(size_t)(v + 8 * half) * OUT_FEATS] = c[v];
  }
}

// ---------------------------------------------------------------------------
// out[n, k, o] = b[o]   (bias is the additive identity of the segment-sum)
// ---------------------------------------------------------------------------
__global__ void __launch_bounds__(256)
efconv_init_out(float* __restrict__ out, const float* __restrict__ b, int total) {
  int i = (int)(blockIdx.x * blockDim.x + threadIdx.x);
  if (i < total) out[i] = b[i & (OUT_FEATS - 1)];
}

// ---------------------------------------------------------------------------
extern "C" void kernel_launch(void* const* d_in, const int* in_sizes, int n_in,
                              void* d_out, int out_size, void* d_ws, size_t ws_size,
                              hipStream_t stream) {
  const float* node_feat = (const float*)d_in[0];   // [N, 64]
  const float* edge_feat = (const float*)d_in[1];   // [E, 8]
  const float* W         = (const float*)d_in[2];   // [64, 64]
  const float* b         = (const float*)d_in[3];   // [64]
  const int*   src       = (const int*)d_in[4];     // [E]
  const int*   dst       = (const int*)d_in[5];     // [E]
  float*       out       = (float*)d_out;           // [N, 512]

  const int n_nodes = in_sizes[0] / IN_FEATS;       // 50000
  const int n_edges = in_sizes[4];                  // 800000

  float* z = (float*)d_ws;                          // [N, 64] = 12.8 MB scratch

  // 1) z = node_feat @ W^T via WMMA (3125 * 4 wave-tiles, 8 waves/block)
  {
    const int tiles  = (n_nodes / 16) * (OUT_FEATS / 16);
    const int blocks = (tiles + 7) / 8;
    efconv_gemm_zw<<<blocks, 256, 0, stream>>>(node_feat, W, z, n_nodes);
  }

  // 2) out = broadcast(b)
  {
    const int total = n_nodes * OUT_STRIDE;
    efconv_init_out<<<(total + 255) / 256, 256, 0, stream>>>(out, b, total);
  }

  // 3) scatter-add edges into out (atomics land in L2: out is 102MB < 192MB)
  {
    efconv_scatter<<<4096, 256, 0, stream>>>(z, edge_feat, src, dst, out, n_edges);
  }
}